// RickerPredation_8839042695687
// MI455X (gfx1250) — compile-verified
//
#include <hip/hip_runtime.h>
#include <hip/hip_bf16.h>
#include <stdint.h>

#define SIGF 0.08f
#define CHUNK 512
#define F_B128_PER_CHUNK (CHUNK * 4 / 512)                       // 4  x b128 (32 lanes x 16B)
#define E_B64_PER_CHUNK  (CHUNK * 8 / 256)                       // 16 x b64  (32 lanes x 8B)
#define ASYNC_PER_CHUNK  (F_B128_PER_CHUNK + E_B64_PER_CHUNK)    // 20 async ops per chunk

// Low 32 bits of a generic pointer to __shared__ data == LDS byte offset
// (flat LDS aperture keeps the DS address in addr[31:0]).
__device__ __forceinline__ uint32_t lds_byte_addr(const void* p) {
    return (uint32_t)(uintptr_t)p;
}

// ---------------------------------------------------------------------------
// Serial scan: one wave, async-LDS double buffering of forcing/eps chunks.
// ---------------------------------------------------------------------------
__global__ __launch_bounds__(32) void ricker_scan_kernel(
    const float* __restrict__ forcing,
    const float* __restrict__ eps,
    const float* __restrict__ init_state,
    const float* __restrict__ params,
    float* __restrict__ out,
    int T)
{
    __shared__ __align__(16) float lds_f[2][CHUNK];
    __shared__ __align__(16) float lds_e[2][2 * CHUNK];

    const int lane = threadIdx.x;
    const int NS = T - 1;                       // number of scan steps
    const int nchunks = (NS + CHUNK - 1) / CHUNK;

    const float a1 = params[0], b1 = params[1], g1 = params[2], bx1 = params[3], cx1 = params[4];
    const float a2 = params[5], b2 = params[6], g2 = params[7], bx2 = params[8], cx2 = params[9];
    const float A11 = a1 * b1, A12 = a1 * g1;   // exp-arg coefficients (off critical path)
    const float A21 = a2 * g2, A22 = a2 * b2;

    const unsigned long long fbase = (unsigned long long)(uintptr_t)forcing;
    const unsigned long long ebase = (unsigned long long)(uintptr_t)eps;
    const uint32_t f_last_b128 = (uint32_t)((size_t)T * 4 - 16);   // last in-bounds b128 start
    const uint32_t e_last_b64  = (uint32_t)((size_t)NS * 8 - 8);   // last in-bounds b64 start

    // Issue one chunk's async global->LDS copies (forcing: b128, eps: b64).
    auto issue = [&](int c, int b) {
        const uint32_t fg0 = (uint32_t)c * (CHUNK * 4);
        const uint32_t eg0 = (uint32_t)c * (CHUNK * 8);
        const uint32_t lf0 = lds_byte_addr(&lds_f[b][0]);
        const uint32_t le0 = lds_byte_addr(&lds_e[b][0]);
#pragma unroll
        for (int j = 0; j < F_B128_PER_CHUNK; ++j) {
            uint32_t go = fg0 + (uint32_t)(j * 512 + lane * 16);
            if (go > f_last_b128) go = f_last_b128;
            const uint32_t lo = lf0 + (uint32_t)(j * 512 + lane * 16);
            asm volatile("global_load_async_to_lds_b128 %0, %1, %2"
                         :: "v"(lo), "v"(go), "s"(fbase) : "memory");
        }
#pragma unroll
        for (int j = 0; j < E_B64_PER_CHUNK; ++j) {
            uint32_t go = eg0 + (uint32_t)(j * 256 + lane * 8);
            if (go > e_last_b64) go = e_last_b64;  // tail clamp hits only never-consumed slot
            const uint32_t lo = le0 + (uint32_t)(j * 256 + lane * 8);
            asm volatile("global_load_async_to_lds_b64 %0, %1, %2"
                         :: "v"(lo), "v"(go), "s"(ebase) : "memory");
        }
    };

    float n1 = init_state[0];
    float n2 = init_state[1];
    float keep1 = n1, keep2 = n2;

    issue(0, 0);   // prologue: start streaming chunk 0

    for (int c = 0; c < nchunks; ++c) {
        const int b = c & 1;
        if (c + 1 < nchunks) {
            issue(c + 1, (c + 1) & 1);                         // prefetch next chunk
            asm volatile("s_wait_asynccnt %0" :: "n"(ASYNC_PER_CHUNK) : "memory"); // chunk c done
        } else {
            asm volatile("s_wait_asynccnt 0" ::: "memory");
        }

        const int base = c * CHUNK;
        const int rem = NS - base;
        const int nsteps = rem < CHUNK ? rem : CHUNK;
        const float* lf = lds_f[b];
        const float* le = lds_e[b];

        // Software-pipelined LDS reads: fetch step i+1 while exp(step i) runs.
        float fa = lf[0], ea0 = le[0], ea1 = le[1];
        for (int i = 0; i < nsteps; ++i) {
            const float f = fa, e0 = ea0, e1 = ea1;
            const int ip = (i + 1 < nsteps) ? (i + 1) : i;
            fa = lf[ip]; ea0 = le[2 * ip]; ea1 = le[2 * ip + 1];

            // z / w terms depend only on staged data -> off the serial critical path
            const float z1 = f * (bx1 + cx1 * f) + SIGF * e0;
            const float z2 = f * (bx2 + cx2 * f) + SIGF * e1;
            const float w1 = a1 + a1 * z1;      // a1*(1+z1)
            const float w2 = a2 + a2 * z2;      // a2*(1+z2)

            // Critical path: 2 FMA -> v_exp_f32 -> mul (two independent chains interleave)
            const float arg1 = w1 - A11 * n1 - A12 * n2;
            const float arg2 = w2 - A22 * n2 - A21 * n1;
            const float n1n = n1 * __expf(arg1);
            const float n2n = n2 * __expf(arg2);

            const int t = base + i;
            const bool mine = ((t & 31) == lane);   // lane t%32 latches this step
            keep1 = mine ? n1n : keep1;
            keep2 = mine ? n2n : keep2;

            if (((t & 31) == 31) || (t == NS - 1)) {
                const int g0 = t & ~31;
                const int s = g0 + lane;
                if (s <= t) {                        // guard the final partial group
                    out[1 + s]     = keep1;          // n1 block: out[0..T-1]
                    out[T + 1 + s] = keep2;          // n2 block: out[T..2T-1]
                }
            }
            n1 = n1n; n2 = n2n;
        }
    }
}

// ---------------------------------------------------------------------------
// Fully parallel Jacobian pass (memory-bound, ~5us at 23.3 TB/s).
// ---------------------------------------------------------------------------
__global__ __launch_bounds__(256) void ricker_jac_kernel(
    const float* __restrict__ forcing,
    const float* __restrict__ eps,
    const float* __restrict__ init_state,
    const float* __restrict__ params,
    float* __restrict__ out,
    int T)
{
    const int NS = T - 1;
    const int t = blockIdx.x * blockDim.x + threadIdx.x;
    if (t >= NS) return;

    const float a1 = params[0], b1 = params[1], g1 = params[2], bx1 = params[3], cx1 = params[4];
    const float a2 = params[5], b2 = params[6], g2 = params[7], bx2 = params[8], cx2 = params[9];

    const float f  = forcing[t];
    const float e0 = eps[2 * t];
    const float e1 = eps[2 * t + 1];
    const float z1 = f * (bx1 + cx1 * f) + SIGF * e0;
    const float z2 = f * (bx2 + cx2 * f) + SIGF * e1;

    const float i0 = init_state[0];
    const float i1 = init_state[1];
    const float p1 = (t == 0) ? i0 : out[t];        // n1[t], written by scan kernel
    const float p2 = (t == 0) ? i1 : out[T + t];    // n2[t]

    const float E1 = __expf(a1 * (1.0f - b1 * p1 - g1 * p2 + z1));
    const float E2 = __expf(a2 * (1.0f - b2 * p2 - g2 * p1 + z2));

    float4 J;
    J.x = E1 * (1.0f - a1 * b1 * p1);
    J.y = -a1 * g1 * p1 * E1;
    J.z = -a2 * g2 * p2 * E2;
    J.w = E2 * (1.0f - a2 * b2 * p2);

    reinterpret_cast<float4*>(out + 2 * (size_t)T)[t] = J;   // jac block, 16B aligned

    if (t == 0) {                 // out[0]=init n1, out[T]=init n2 (no other thread reads these)
        out[0] = i0;
        out[T] = i1;
    }
}

extern "C" void kernel_launch(void* const* d_in, const int* in_sizes, int n_in,
                              void* d_out, int out_size, void* d_ws, size_t ws_size,
                              hipStream_t stream) {
    (void)n_in; (void)out_size; (void)d_ws; (void)ws_size;
    const float* forcing = (const float*)d_in[0];
    const float* eps     = (const float*)d_in[1];
    const float* init    = (const float*)d_in[2];
    const float* params  = (const float*)d_in[3];
    float* out = (float*)d_out;
    const int T = in_sizes[0];

    ricker_scan_kernel<<<1, 32, 0, stream>>>(forcing, eps, init, params, out, T);

    const int NS = T - 1;
    const int blocks = (NS + 255) / 256;
    ricker_jac_kernel<<<blocks, 256, 0, stream>>>(forcing, eps, init, params, out, T);
}